// LossCDF_51350628991247
// MI455X (gfx1250) — compile-verified
//
#include <hip/hip_runtime.h>
#include <stdint.h>

#define EPS   0.001f
#define NBINS 128
// ws layout (floats): [0,129) = e_t edges ; pad [129] ;
//                     [130 + 2i] = slope[i], [131 + 2i] = bias[i]  (i < 128)
#define WS_SB 130
#define TAB_N (WS_SB + 2 * NBINS)   // 386 floats

typedef __attribute__((ext_vector_type(4))) float v4f;
typedef __attribute__((ext_vector_type(2))) float v2f;

// ---------------------------------------------------------------------------
// Hot kernel first (so its disasm shows in the snippet).
// Streaming transform: HBM-bound (~256 MB total -> ~11 us ideal @ 23.3 TB/s).
// Table staged via gfx1250 async global->LDS (ASYNCcnt). Per element:
//   coarse level in registers (7 compares) + 4 LDS binary steps + 1 b64 pair.
// ---------------------------------------------------------------------------
__global__ __launch_bounds__(256) void lcdf_map(const float* __restrict__ t,
                                                const float* __restrict__ ws,
                                                float* __restrict__ out,
                                                int n) {
  __shared__ __align__(16) float tab[TAB_N];

#if __has_builtin(__builtin_amdgcn_global_load_async_to_lds_b32)
  // CDNA5 async data mover path: per-lane global->LDS, tracked by ASYNCcnt.
  for (int i = threadIdx.x; i < TAB_N; i += blockDim.x) {
    __builtin_amdgcn_global_load_async_to_lds_b32(
        (int*)(ws + i), (int*)(tab + i), 0, 0);
  }
  __asm__ volatile("s_wait_asynccnt 0" ::: "memory");
#else
  for (int i = threadIdx.x; i < TAB_N; i += blockDim.x) tab[i] = ws[i];
#endif
  __syncthreads();

  const float* __restrict__ et = tab;                      // 129 edges
  const v2f*   __restrict__ sb = (const v2f*)(tab + WS_SB); // (slope,bias) pairs

  // Coarse edges held in VGPRs: replaces 3 LDS search levels with VALU.
  const float c16 = et[16],  c32 = et[32],  c48 = et[48], c64 = et[64];
  const float c80 = et[80],  c96 = et[96],  c112 = et[112];

  const int n4  = n >> 2;
  const int gsz = (int)(gridDim.x * blockDim.x);
  const int gid = (int)(blockIdx.x * blockDim.x + threadIdx.x);

  // vectorized main loop: B128 NT load, 4x branchless search, B128 NT store
  for (int i = gid; i < n4; i += gsz) {
    v4f v = __builtin_nontemporal_load((const v4f*)t + i);
    v4f r;
#pragma unroll
    for (int c = 0; c < 4; ++c) {
      const float x = v[c];
      // coarse: largest multiple-of-16 idx (<=112) with et[idx] <= x
      int idx = ((int)(c16 <= x) + (int)(c32 <= x) + (int)(c48 <= x) +
                 (int)(c64 <= x) + (int)(c80 <= x) + (int)(c96 <= x) +
                 (int)(c112 <= x)) << 4;
      // refine within the 16-wide segment; final idx in [0,127]
#pragma unroll
      for (int s = 8; s >= 1; s >>= 1) {
        const int p = idx + s;
        idx = (et[p] <= x) ? p : idx;
      }
      const v2f pr = sb[idx];              // one ds_load_b64: (slope, bias)
      r[c] = fmaf(x, pr.x, pr.y);
    }
    __builtin_nontemporal_store(r, (v4f*)out + i);
  }

  // scalar tail (not hit for 8192x4096 but kept general)
  for (int i = (n & ~3) + gid; i < n; i += gsz) {
    const float x = t[i];
    int idx = ((int)(c16 <= x) + (int)(c32 <= x) + (int)(c48 <= x) +
               (int)(c64 <= x) + (int)(c80 <= x) + (int)(c96 <= x) +
               (int)(c112 <= x)) << 4;
#pragma unroll
    for (int s = 8; s >= 1; s >>= 1) {
      const int p = idx + s;
      idx = (et[p] <= x) ? p : idx;
    }
    const v2f pr = sb[idx];
    out[i] = fmaf(x, pr.x, pr.y);
  }
}

// ---------------------------------------------------------------------------
// Kernel 2: one block, 128 threads. Builds e_t edges and per-bin slope/bias.
// O(128) work — negligible; accuracy over speed here.
// ---------------------------------------------------------------------------
__global__ void lcdf_build_tables(const float* __restrict__ l_t,
                                  const float* __restrict__ l_u,
                                  float* __restrict__ ws) {
  __shared__ float swt[NBINS], swu[NBINS], set[NBINS + 1], seu[NBINS + 1];
  const int i = threadIdx.x;  // 0..127
  swt[i] = l_t[i];
  swu[i] = l_u[i];
  __syncthreads();
  if (i == 0) {
    // softmax(l_t) with max-subtraction; exp(l_u)
    float m = swt[0];
    for (int k = 1; k < NBINS; ++k) m = fmaxf(m, swt[k]);
    float st = 0.f, su = 0.f;
    for (int k = 0; k < NBINS; ++k) {
      swt[k] = expf(swt[k] - m); st += swt[k];
      swu[k] = expf(swu[k]);     su += swu[k];
    }
    const float inv_st   = 1.f / st;
    const float renorm_t = 1.f / (1.f + (float)NBINS * EPS);  // sum(softmax+eps)
    const float inv_su   = 1.f / (su + (float)NBINS * EPS);   // sum(exp+eps)
    float at = 0.f, au = 0.f;
    set[0] = 0.f; seu[0] = 0.f;
    for (int k = 0; k < NBINS; ++k) {
      at += (swt[k] * inv_st + EPS) * renorm_t;  // w_t cumsum
      au += (swu[k] + EPS) * inv_su;             // w_u cumsum
      set[k + 1] = at;
      seu[k + 1] = au;
    }
  }
  __syncthreads();
  const float et0 = set[i], et1 = set[i + 1];
  const float eu0 = seu[i], eu1 = seu[i + 1];
  const float slope = (eu1 - eu0) / (et1 - et0);   // division hoisted out of hot loop
  ws[i] = et0;
  if (i == 0) {
    ws[NBINS] = set[NBINS];   // e_t[128]
    ws[NBINS + 1] = 0.f;      // pad slot: keep staged LDS fully deterministic
  }
  ws[WS_SB + 2 * i]     = slope;
  ws[WS_SB + 2 * i + 1] = fmaf(-slope, et0, eu0);  // u = bias + slope * t
}

// ---------------------------------------------------------------------------
extern "C" void kernel_launch(void* const* d_in, const int* in_sizes, int n_in,
                              void* d_out, int out_size, void* d_ws, size_t ws_size,
                              hipStream_t stream) {
  const float* t  = (const float*)d_in[0];
  const float* lt = (const float*)d_in[1];
  const float* lu = (const float*)d_in[2];
  float* out = (float*)d_out;
  float* ws  = (float*)d_ws;
  const int n = in_sizes[0];

  lcdf_build_tables<<<1, NBINS, 0, stream>>>(lt, lu, ws);

  const int threads = 256;                     // 8 wave32 waves per block
  const long long n4 = (long long)(n >> 2);
  long long work = n4 > 0 ? n4 : (long long)n;
  long long blocks_ll = (work + (long long)threads * 8 - 1) / ((long long)threads * 8);
  if (blocks_ll < 1) blocks_ll = 1;
  if (blocks_ll > (1 << 20)) blocks_ll = (1 << 20);
  lcdf_map<<<(int)blocks_ll, threads, 0, stream>>>(t, ws, out, n);
}